// GNNStack_9594956939372
// MI455X (gfx1250) — compile-verified
//
#include <hip/hip_runtime.h>

typedef __attribute__((ext_vector_type(2))) float v2f;
typedef __attribute__((ext_vector_type(8))) float v8f;

#define HID 128
#define LN_EPS 1e-5f

// ---------------- utility kernels ----------------

__global__ void zero_kernel(float* __restrict__ p, long long n) {
    long long idx = ((long long)blockIdx.x * blockDim.x + threadIdx.x) * 4;
    if (idx >= n) return;
    if (idx + 4 <= n) {
        float4 z = {0.f, 0.f, 0.f, 0.f};
        *(float4*)(p + idx) = z;
    } else {
        for (long long j = idx; j < n; ++j) p[j] = 0.f;
    }
}

__global__ void deg_kernel(const int* __restrict__ col, float* __restrict__ deg, int E) {
    int e = blockIdx.x * blockDim.x + threadIdx.x;
    if (e < E) atomicAdd(&deg[col[e]], 1.0f);
}

// dinv = rsqrt(deg + 1)  (the +1 is the self-loop; deg>0 always holds)
__global__ void dinv_kernel(float* __restrict__ deg, int N) {
    int i = blockIdx.x * blockDim.x + threadIdx.x;
    if (i < N) deg[i] = rsqrtf(deg[i] + 1.0f);
}

// x[i,:] = emb_table[node_type[i],:]  (one thread per 4 features)
__global__ void embed_kernel(const int* __restrict__ node_type,
                             const float* __restrict__ emb,
                             float* __restrict__ x, int N) {
    long long tid = (long long)blockIdx.x * blockDim.x + threadIdx.x;
    int i = (int)(tid >> 5);
    int f = ((int)tid & 31) * 4;
    if (i >= N) return;
    int t = node_type[i];
    *(float4*)(x + (size_t)i * HID + f) = *(const float4*)(emb + (size_t)t * HID + f);
}

// ---------------- fp32 WMMA GEMM: C[M,128] = A[M,128] @ B[128,128] ----------------
// One wave computes a 16-row x 128-col strip (8 v8f accumulators).
// B is staged into LDS pre-packed in WMMA-fragment order so the inner loop
// reads each lane's 8 fragments with 4x ds_load_b128 (bank-conflict-free via
// 20-float row stride) instead of 16 scalar global loads.
// K is staged in two 64-wide halves -> 40KB static LDS per block.
#define BST 20                       // padded stride (floats) per (kk,hi,l15) row
#define B_LDS_FLOATS (16 * 2 * 16 * BST)   // 10240 floats = 40KB

template <int FUSE>
__global__ void gemm128_wmma(const float* __restrict__ A, const float* __restrict__ B,
                             const float* __restrict__ bias, float* __restrict__ C,
                             int M) {
    __shared__ float Bs[B_LDS_FLOATS];

    int tid  = threadIdx.x;
    int wave = (int)(((long long)blockIdx.x * blockDim.x + tid) >> 5);
    int lane = tid & 31;
    int m0   = wave * 16;
    bool active = (m0 < M);
    int l15 = lane & 15;
    int hi  = lane >> 4;                 // 0: K=k,k+1   1: K=k+2,k+3

    v8f acc[8] = {};
    const float* Arow = A + (size_t)((active ? m0 : 0) + l15) * HID;

    for (int kb = 0; kb < 2; ++kb) {     // two 64-wide K halves
        // ---- cooperative stage of B half into fragment-packed LDS ----
        // pairs: (kk:16) x (hi:2) x (l15:16) x (j:8) -> 4096, 2 floats each
        for (int idx = tid; idx < 4096; idx += 256) {
            int j   = idx & 7;
            int sl  = (idx >> 3) & 15;
            int sh  = (idx >> 7) & 1;
            int kk  = idx >> 8;
            int k   = kb * 64 + kk * 4 + 2 * sh;
            int n   = j * 16 + sl;
            int dst = ((kk * 2 + sh) * 16 + sl) * BST + j * 2;
            Bs[dst]     = B[(size_t)k * HID + n];
            Bs[dst + 1] = B[(size_t)(k + 1) * HID + n];
        }
        __syncthreads();

        if (active) {
            for (int kk = 0; kk < 16; ++kk) {
                int k = kb * 64 + kk * 4;
                v2f a = *(const v2f*)(Arow + k + 2 * hi);
                const float* bp = &Bs[((kk * 2 + hi) * 16 + l15) * BST];
                float4 q0 = *(const float4*)(bp + 0);    // j=0,1
                float4 q1 = *(const float4*)(bp + 4);    // j=2,3
                float4 q2 = *(const float4*)(bp + 8);    // j=4,5
                float4 q3 = *(const float4*)(bp + 12);   // j=6,7
                v2f b0 = {q0.x, q0.y}, b1 = {q0.z, q0.w};
                v2f b2 = {q1.x, q1.y}, b3 = {q1.z, q1.w};
                v2f b4 = {q2.x, q2.y}, b5 = {q2.z, q2.w};
                v2f b6 = {q3.x, q3.y}, b7 = {q3.z, q3.w};
                acc[0] = __builtin_amdgcn_wmma_f32_16x16x4_f32(false, a, false, b0, (short)0, acc[0], false, false);
                acc[1] = __builtin_amdgcn_wmma_f32_16x16x4_f32(false, a, false, b1, (short)0, acc[1], false, false);
                acc[2] = __builtin_amdgcn_wmma_f32_16x16x4_f32(false, a, false, b2, (short)0, acc[2], false, false);
                acc[3] = __builtin_amdgcn_wmma_f32_16x16x4_f32(false, a, false, b3, (short)0, acc[3], false, false);
                acc[4] = __builtin_amdgcn_wmma_f32_16x16x4_f32(false, a, false, b4, (short)0, acc[4], false, false);
                acc[5] = __builtin_amdgcn_wmma_f32_16x16x4_f32(false, a, false, b5, (short)0, acc[5], false, false);
                acc[6] = __builtin_amdgcn_wmma_f32_16x16x4_f32(false, a, false, b6, (short)0, acc[6], false, false);
                acc[7] = __builtin_amdgcn_wmma_f32_16x16x4_f32(false, a, false, b7, (short)0, acc[7], false, false);
            }
        }
        __syncthreads();   // safe to restage (all waves done reading)
    }

    if (!active) return;

#pragma unroll
    for (int j = 0; j < 8; ++j) {
        int n = j * 16 + l15;
        float bv = FUSE ? bias[n] : 0.f;
#pragma unroll
        for (int v = 0; v < 8; ++v) {
            int m = m0 + v + 8 * hi;
            float val = acc[j][v];
            if (FUSE) val = fmaxf(val + bv, 0.f);
            C[(size_t)m * HID + n] = val;
        }
    }
}

// ---------------- edge aggregation: conv[row[e]] += hW[col[e]] * dinv[row]*dinv[col] ----------------
// One wave per edge, 4 floats per lane.
__global__ void aggregate_kernel(const float* __restrict__ hW,
                                 const int* __restrict__ row,
                                 const int* __restrict__ col,
                                 const float* __restrict__ dinv,
                                 float* __restrict__ conv, int E) {
    long long tid = (long long)blockIdx.x * blockDim.x + threadIdx.x;
    int e = (int)(tid >> 5);
    int f = ((int)tid & 31) * 4;
    if (e >= E) return;
    int r = row[e], c = col[e];
    float nrm = dinv[r] * dinv[c];
    float4 v = *(const float4*)(hW + (size_t)c * HID + f);
    float* dst = conv + (size_t)r * HID + f;
    atomicAdd(dst + 0, v.x * nrm);
    atomicAdd(dst + 1, v.y * nrm);
    atomicAdd(dst + 2, v.z * nrm);
    atomicAdd(dst + 3, v.w * nrm);
}

// ---------------- fused skip + self-loop + bias + (emb store) + ReLU + LayerNorm ----------------
// One wave per node row (128 features = 4/lane, wave32 shuffle reduction).
__global__ void combine_kernel(float* __restrict__ x,
                               const float* __restrict__ hW,
                               const float* __restrict__ conv,
                               const float* __restrict__ dinv,
                               const float* __restrict__ bias,
                               const float* __restrict__ lnscale,
                               const float* __restrict__ lnbias,
                               float* __restrict__ emb_out,
                               int N, int do_ln, int store_emb) {
    long long tid = (long long)blockIdx.x * blockDim.x + threadIdx.x;
    int i = (int)(tid >> 5);
    int lane = threadIdx.x & 31;
    if (i >= N) return;
    size_t base = (size_t)i * HID;
    float di = dinv[i];
    float selfn = di * di;

    float v[4];
#pragma unroll
    for (int t = 0; t < 4; ++t) {
        int f = lane + t * 32;
        v[t] = x[base + f] + conv[base + f] + selfn * hW[base + f] + bias[f];
    }
    if (store_emb) {
#pragma unroll
        for (int t = 0; t < 4; ++t) emb_out[base + lane + t * 32] = v[t];
    }
#pragma unroll
    for (int t = 0; t < 4; ++t) v[t] = fmaxf(v[t], 0.f);

    if (do_ln) {
        float s = v[0] + v[1] + v[2] + v[3];
        float q = v[0] * v[0] + v[1] * v[1] + v[2] * v[2] + v[3] * v[3];
#pragma unroll
        for (int off = 16; off > 0; off >>= 1) {
            s += __shfl_xor(s, off, 32);
            q += __shfl_xor(q, off, 32);
        }
        float mu = s * (1.0f / HID);
        float var = q * (1.0f / HID) - mu * mu;
        float inv = rsqrtf(var + LN_EPS);
#pragma unroll
        for (int t = 0; t < 4; ++t) {
            int f = lane + t * 32;
            v[t] = (v[t] - mu) * inv * lnscale[f] + lnbias[f];
        }
    }
#pragma unroll
    for (int t = 0; t < 4; ++t) x[base + lane + t * 32] = v[t];
}

// h0[p,:] = x[premise_index[p],:]
__global__ void gather_kernel(const float* __restrict__ x,
                              const int* __restrict__ idx,
                              float* __restrict__ h, int P) {
    long long tid = (long long)blockIdx.x * blockDim.x + threadIdx.x;
    int p = (int)(tid >> 5);
    int f = ((int)tid & 31) * 4;
    if (p >= P) return;
    int n = idx[p];
    *(float4*)(h + (size_t)p * HID + f) = *(const float4*)(x + (size_t)n * HID + f);
}

// score[p] = dot(h[p,:], W_out) + b_out   (one wave per row)
__global__ void score_kernel(const float* __restrict__ h,
                             const float* __restrict__ Wout,
                             const float* __restrict__ bout,
                             float* __restrict__ out, int P) {
    long long tid = (long long)blockIdx.x * blockDim.x + threadIdx.x;
    int p = (int)(tid >> 5);
    int lane = threadIdx.x & 31;
    if (p >= P) return;
    const float* hr = h + (size_t)p * HID;
    float s = 0.f;
#pragma unroll
    for (int t = 0; t < 4; ++t) {
        int f = lane + t * 32;
        s += hr[f] * Wout[f];
    }
#pragma unroll
    for (int off = 16; off > 0; off >>= 1) s += __shfl_xor(s, off, 32);
    if (lane == 0) out[p] = s + bout[0];
}

// ---------------- launch ----------------

extern "C" void kernel_launch(void* const* d_in, const int* in_sizes, int n_in,
                              void* d_out, int out_size, void* d_ws, size_t ws_size,
                              hipStream_t stream) {
    const int N = 100000, E = 600000, P = 20000;

    const int*   node_type     = (const int*)d_in[0];
    const int*   edge_index    = (const int*)d_in[1];
    const int*   premise_index = (const int*)d_in[2];
    const float* emb_table     = (const float*)d_in[3];
    const float* Ws            = (const float*)d_in[4];
    const float* bs            = (const float*)d_in[5];
    const float* ln_scale      = (const float*)d_in[6];
    const float* ln_bias       = (const float*)d_in[7];
    const float* W_dense       = (const float*)d_in[8];
    const float* b_dense       = (const float*)d_in[9];
    const float* W_out         = (const float*)d_in[10];
    const float* b_out         = (const float*)d_in[11];

    float* out_emb   = (float*)d_out;                      // N*HID
    float* out_score = (float*)d_out + (size_t)N * HID;    // P

    float* x    = (float*)d_ws;
    float* hW   = x    + (size_t)N * HID;
    float* conv = hW   + (size_t)N * HID;
    float* dinv = conv + (size_t)N * HID;
    float* h0   = dinv + (((size_t)N + 255) & ~(size_t)255);
    float* h1   = h0   + (size_t)P * HID;

    const int* row = edge_index;        // edge_index[0]
    const int* col = edge_index + E;    // edge_index[1]

    const int TB = 256;
    long long nodeFeat = (long long)N * HID;

    // gcn_norm degrees
    zero_kernel<<<(int)((N / 4 + TB - 1) / TB), TB, 0, stream>>>(dinv, N);
    deg_kernel<<<(E + TB - 1) / TB, TB, 0, stream>>>(col, dinv, E);
    dinv_kernel<<<(N + TB - 1) / TB, TB, 0, stream>>>(dinv, N);

    // x = emb_table[node_type]
    embed_kernel<<<(int)(((long long)N * 32 + TB - 1) / TB), TB, 0, stream>>>(
        node_type, emb_table, x, N);

    int gemm_blocks_N = ((N + 15) / 16 + 7) / 8;   // 8 waves/block
    int waveN_blocks  = (int)(((long long)N * 32 + TB - 1) / TB);

    for (int layer = 0; layer < 4; ++layer) {
        // hW = x @ W[layer]
        gemm128_wmma<0><<<gemm_blocks_N, TB, 0, stream>>>(
            x, Ws + (size_t)layer * HID * HID, nullptr, hW, N);
        // conv = 0
        zero_kernel<<<(int)((nodeFeat / 4 + TB - 1) / TB), TB, 0, stream>>>(conv, nodeFeat);
        // conv[row] += hW[col] * norm  (non-self-loop edges)
        aggregate_kernel<<<(int)(((long long)E * 32 + TB - 1) / TB), TB, 0, stream>>>(
            hW, row, col, dinv, conv, E);
        // x = x + conv + dinv^2*hW + b ; emb ; relu ; LN
        int ln_idx = layer < 3 ? layer : 2;  // pointer kept in range; unused when do_ln=0
        combine_kernel<<<waveN_blocks, TB, 0, stream>>>(
            x, hW, conv, dinv,
            bs + (size_t)layer * HID,
            ln_scale + (size_t)ln_idx * HID,
            ln_bias + (size_t)ln_idx * HID,
            out_emb, N, layer < 3 ? 1 : 0, layer == 3 ? 1 : 0);
    }

    // readout
    gather_kernel<<<(int)(((long long)P * 32 + TB - 1) / TB), TB, 0, stream>>>(
        x, premise_index, h0, P);

    int gemm_blocks_P = ((P + 15) / 16 + 7) / 8;
    gemm128_wmma<1><<<gemm_blocks_P, TB, 0, stream>>>(
        h0, W_dense, b_dense, h1, P);
    gemm128_wmma<1><<<gemm_blocks_P, TB, 0, stream>>>(
        h1, W_dense + (size_t)HID * HID, b_dense + HID, h0, P);

    score_kernel<<<(int)(((long long)P * 32 + TB - 1) / TB), TB, 0, stream>>>(
        h0, W_out, b_out, out_score, P);
}